// ExtractTensorPatches_18992345383471
// MI455X (gfx1250) — compile-verified
//
#include <hip/hip_runtime.h>
#include <stdint.h>

// ExtractTensorPatches: 3x3 unfold, stride 1, pad 1, fp32.
// x: (B=8, C=32, H=256, W=256). Output flat layout == patches[b][c][k][h][w].
// Pure data movement: 64 MB in, 576 MB out -> HBM store-bound (~27us floor
// at 23.3 TB/s). Strategy: async-DMA each halo tile into LDS once, emit the
// 9 shifted copies as non-temporal b128 stores.

typedef float vfloat4 __attribute__((ext_vector_type(4)));
typedef int   vint4   __attribute__((ext_vector_type(4)));

#define B_      8
#define C_      32
#define H_      256
#define W_      256
#define TILE_H  16
#define LDSROWS (TILE_H + 2)     // 18 rows (1-row halo top+bottom)
#define LDSW    264              // row stride (floats): 1056 B, multiple of 16
#define COL0    4                // interior start col: global col g -> lds col g+4
                                 // (keeps every 16B async chunk 16B-aligned in LDS)

#if defined(__has_builtin)
#  if __has_builtin(__builtin_amdgcn_global_load_async_to_lds_b128)
#    define HAVE_ASYNC_B128 1
#  endif
#  if __has_builtin(__builtin_amdgcn_s_wait_asynccnt)
#    define HAVE_WAIT_ASYNC 1
#  endif
#endif

// Builtin signature: (v4i addrspace(1)*, v4i addrspace(3)*, imm offset, imm cpol)
typedef __attribute__((address_space(1))) vint4 g_vint4;
typedef __attribute__((address_space(3))) vint4 l_vint4;

__global__ __launch_bounds__(256)
void ExtractTensorPatches_unfold3x3(const float* __restrict__ x,
                                    float* __restrict__ out) {
    __shared__ float tile[LDSROWS * LDSW];   // 19,008 bytes

    const int tid = threadIdx.x;
    const int bid = blockIdx.x;
    const int t   = bid & 15;                // h-tile index (H / TILE_H = 16)
    const int pl  = bid >> 4;                // plane = b*C + c
    const int h0  = t * TILE_H;

    const float* src = x + ((size_t)pl << 16);          // pl * H*W

    // ---- 1) zero the whole tile: provides the zero-pad halo for free ----
    for (int i = tid; i < LDSROWS * LDSW; i += 256) tile[i] = 0.0f;
    __syncthreads();

    // ---- 2) async DMA the interior: 18 lds rows x 64 chunks of 16B ----
    for (int i = tid; i < LDSROWS * (W_ / 4); i += 256) {
        const int r  = i >> 6;               // lds row 0..17
        const int ck = i & 63;               // 16B chunk within the row
        const int gr = h0 + r - 1;           // global row (pad offset -1)
        if (0 <= gr && gr < H_) {
            const float* g = src + gr * W_ + ck * 4;
            float*       l = &tile[r * LDSW + COL0 + ck * 4];
#if HAVE_ASYNC_B128
            __builtin_amdgcn_global_load_async_to_lds_b128(
                (g_vint4*)(uintptr_t)g,
                (l_vint4*)(uint32_t)(uintptr_t)l,
                /*offset=*/0, /*cpol=*/0);
#else
            *(vfloat4*)l = *(const vfloat4*)g;   // sync fallback
#endif
        }
    }
#if HAVE_ASYNC_B128
#  if HAVE_WAIT_ASYNC
    __builtin_amdgcn_s_wait_asynccnt(0);
#  else
    asm volatile("s_wait_asynccnt 0" ::: "memory");
#  endif
#endif
    __syncthreads();

    // ---- 3) emit 9 shifted copies; 4 output cols (one b128) per store ----
    // out flat index = (((b*C + c)*9 + k)*H + h)*W + w  ==  pl*9*65536 + ...
    float* outp = out + (size_t)pl * (9u * H_ * W_);

    for (int i = tid; i < TILE_H * (W_ / 4); i += 256) {
        const int hr = i >> 6;               // output row within tile
        const int cw = i & 63;               // float4 chunk (output cols 4cw..4cw+3)

        // Window of input values needed for all 9 taps of this chunk:
        // rows hr..hr+2 (lds), lds cols 4cw+3 .. 4cw+8
        float v[3][6];
#pragma unroll
        for (int rr = 0; rr < 3; ++rr) {
#pragma unroll
            for (int cc = 0; cc < 6; ++cc)
                v[rr][cc] = tile[(hr + rr) * LDSW + 4 * cw + (COL0 - 1) + cc];
        }

        const size_t rowoff = (size_t)(h0 + hr) * W_ + 4 * cw;
#pragma unroll
        for (int ki = 0; ki < 3; ++ki) {
#pragma unroll
            for (int kj = 0; kj < 3; ++kj) {
                vfloat4 o = { v[ki][kj], v[ki][kj + 1], v[ki][kj + 2], v[ki][kj + 3] };
                float* dst = outp + (size_t)(ki * 3 + kj) * (H_ * W_) + rowoff;
                __builtin_nontemporal_store(o, (vfloat4*)dst);
            }
        }
    }
}

extern "C" void kernel_launch(void* const* d_in, const int* in_sizes, int n_in,
                              void* d_out, int out_size, void* d_ws, size_t ws_size,
                              hipStream_t stream) {
    const float* x = (const float*)d_in[0];
    float* out     = (float*)d_out;
    // B*C*(H/TILE_H) = 8*32*16 = 4096 blocks, 256 threads (8 wave32)
    ExtractTensorPatches_unfold3x3<<<dim3(4096), dim3(256), 0, stream>>>(x, out);
}